// SPAligned_26774826123435
// MI455X (gfx1250) — compile-verified
//
#include <hip/hip_runtime.h>
#include <math.h>

typedef float v2f __attribute__((ext_vector_type(2)));
typedef float v8f __attribute__((ext_vector_type(8)));

#define HW   3136
#define WID  56
#define HEI  56
#define CH   256
#define CQ   64
#define NKEY 50   // 1 self + 49 neighborhood

__device__ __forceinline__ v8f wmma4(v2f a, v2f b, v8f c) {
  // D = A(16x4 f32) * B(4x16 f32) + C(16x16 f32)
  return __builtin_amdgcn_wmma_f32_16x16x4_f32(
      /*neg_a=*/false, a, /*neg_b=*/false, b,
      /*c_mod=*/(short)0, c, /*reuse_a=*/false, /*reuse_b=*/false);
}

// -------------------------------------------------------------------------
// Generic f32 WMMA GEMM:  C[M,N] = A[M,K] * B[K,N]   (row-major, N == HW)
// One wave computes a 16(M) x 64(N) tile with 4 accumulators.
// blockIdx.z selects the image (B/C advance by per-image strides).
// -------------------------------------------------------------------------
__global__ __launch_bounds__(256)
void gemm_wmma(const float* __restrict__ A, const float* __restrict__ B,
               float* __restrict__ Cc, int M, int K,
               long bImgStride, long cImgStride) {
  const int N = HW;
  B  += (long)blockIdx.z * bImgStride;
  Cc += (long)blockIdx.z * cImgStride;
  const int lane = threadIdx.x & 31;
  const int wave = (blockIdx.x * blockDim.x + threadIdx.x) >> 5;
  const int tilesN = N / 64;
  const int tilesM = M / 16;
  if (wave >= tilesM * tilesN) return;
  const int tm = wave / tilesN, tn = wave % tilesN;
  const int m0 = tm * 16, n0 = tn * 64;
  const int hh = lane >> 4;    // half of wave -> which K pair / which M group
  const int ml = lane & 15;    // row (A) / column (B)

  v8f acc0 = {0}, acc1 = {0}, acc2 = {0}, acc3 = {0};
  const float* Arow = A + (long)(m0 + ml) * K + 2 * hh;
  for (int kb = 0; kb < K; kb += 4) {
    v2f a; a.x = Arow[kb]; a.y = Arow[kb + 1];
    const float* B0 = B + (long)(kb + 2 * hh) * N + n0 + ml;
    const float* B1 = B0 + N;
    v2f b0 = { B0[0],  B1[0]  };
    v2f b1 = { B0[16], B1[16] };
    v2f b2 = { B0[32], B1[32] };
    v2f b3 = { B0[48], B1[48] };
    acc0 = wmma4(a, b0, acc0);
    acc1 = wmma4(a, b1, acc1);
    acc2 = wmma4(a, b2, acc2);
    acc3 = wmma4(a, b3, acc3);
  }
#pragma unroll
  for (int r = 0; r < 8; ++r) {
    const int m = m0 + r + 8 * hh;
    float* Cp = Cc + (long)m * N + n0 + ml;
    Cp[0]  = acc0[r];
    Cp[16] = acc1[r];
    Cp[32] = acc2[r];
    Cp[48] = acc3[r];
  }
}

// -------------------------------------------------------------------------
// Attention: one wave per pixel. Logits from precomputed Q/Kself/Kmap
// (OOB neighbor => projected-zero key => logit exactly 0, matching the
// zero-padded reference). Softmax over 50 (wave32 shuffle reductions),
// weighted sum over the 49 neighborhood raw tb vectors (self column dropped).
// -------------------------------------------------------------------------
__global__ __launch_bounds__(256)
void attn_kernel(const float* __restrict__ Q, const float* __restrict__ Ksf,
                 const float* __restrict__ Km, const float* __restrict__ TB,
                 float* __restrict__ OUT, long qImgStride, long tImgStride) {
  const float* Qi  = Q   + (long)blockIdx.z * qImgStride;
  const float* Ki  = Ksf + (long)blockIdx.z * qImgStride;
  const float* Kmi = Km  + (long)blockIdx.z * qImgStride;
  const float* tb  = TB  + (long)blockIdx.z * tImgStride;
  float*       out = OUT + (long)blockIdx.z * tImgStride;

  const int lane = threadIdx.x & 31;
  const int wave = threadIdx.x >> 5;
  const int n = blockIdx.x * 8 + wave;      // pixel index (uniform per wave)
  if (n >= HW) return;
  const int py = n / WID, px = n % WID;

  // ---- logits: lane handles key indices jj = lane and lane+32 ----
  float lg0 = 0.0f, lg1;
  const bool v1 = (lane + 32) < NKEY;
  {
    const int jj = lane;
    const float* kvec = Ki + n;             // self key default
    bool inb = true;
    if (jj != 0) {
      const int j = jj - 1;
      const int yy = py + j / 7 - 3, xx = px + j % 7 - 3;
      inb = (yy >= 0 && yy < HEI && xx >= 0 && xx < WID);
      kvec = Kmi + (yy * WID + xx);
    }
    if (inb) {
      float s = 0.f;
      for (int d = 0; d < CQ; ++d) s += Qi[(long)d * HW + n] * kvec[(long)d * HW];
      lg0 = s;
    }
  }
  if (v1) {
    const int j = lane + 31;                // (lane+32) - 1
    const int yy = py + j / 7 - 3, xx = px + j % 7 - 3;
    lg1 = 0.0f;
    if (yy >= 0 && yy < HEI && xx >= 0 && xx < WID) {
      const float* kvec = Kmi + (yy * WID + xx);
      float s = 0.f;
      for (int d = 0; d < CQ; ++d) s += Qi[(long)d * HW + n] * kvec[(long)d * HW];
      lg1 = s;
    }
  } else {
    lg1 = -__builtin_inff();
  }

  // ---- softmax over all 50 (reference drops column 0 after softmax) ----
  float mx = fmaxf(lg0, lg1);
  for (int off = 16; off > 0; off >>= 1) mx = fmaxf(mx, __shfl_xor(mx, off, 32));
  const float e0 = expf(lg0 - mx);
  const float e1 = v1 ? expf(lg1 - mx) : 0.f;
  float sum = e0 + e1;
  for (int off = 16; off > 0; off >>= 1) sum += __shfl_xor(sum, off, 32);
  const float w0 = e0 / sum, w1 = e1 / sum;

  // ---- weighted sum over keys 1..49; each lane owns 8 channels ----
  float o[8] = {0, 0, 0, 0, 0, 0, 0, 0};
  for (int j = 1; j < NKEY; ++j) {
    const float w = (j < 32) ? __shfl(w0, j, 32) : __shfl(w1, j - 32, 32);
    const int jm1 = j - 1;
    const int yy = py + jm1 / 7 - 3, xx = px + jm1 % 7 - 3;
    if (yy >= 0 && yy < HEI && xx >= 0 && xx < WID) {
      const float* p = tb + (yy * WID + xx);
#pragma unroll
      for (int c = 0; c < 8; ++c) o[c] += w * p[(long)(lane * 8 + c) * HW];
    }
  }
#pragma unroll
  for (int c = 0; c < 8; ++c) out[(long)(lane * 8 + c) * HW + n] = o[c];
}

// -------------------------------------------------------------------------
// Final: Y = Wv[256,512] * [x;xt][512,HW] per image, fused +bv, BN, ReLU.
// The 512-row B matrix is virtually concatenated via pointer select.
// -------------------------------------------------------------------------
__global__ __launch_bounds__(256)
void final_wmma(const float* __restrict__ Wv, const float* __restrict__ X,
                const float* __restrict__ XT, const float* __restrict__ bv,
                const float* __restrict__ gamma, const float* __restrict__ beta,
                const float* __restrict__ rmean, const float* __restrict__ rvar,
                float* __restrict__ Y) {
  const int N = HW, K = 2 * CH, M = CH;
  const long img = (long)blockIdx.z * CH * HW;
  const float* x  = X  + img;
  const float* xt = XT + img;
  float*       y  = Y  + img;
  const int lane = threadIdx.x & 31;
  const int wave = (blockIdx.x * blockDim.x + threadIdx.x) >> 5;
  const int tilesN = N / 64;   // 49
  const int tilesM = M / 16;   // 16
  if (wave >= tilesM * tilesN) return;
  const int tm = wave / tilesN, tn = wave % tilesN;
  const int m0 = tm * 16, n0 = tn * 64;
  const int hh = lane >> 4, ml = lane & 15;

  v8f acc0 = {0}, acc1 = {0}, acc2 = {0}, acc3 = {0};
  const float* Arow = Wv + (long)(m0 + ml) * K + 2 * hh;
  for (int kb = 0; kb < K; kb += 4) {
    v2f a; a.x = Arow[kb]; a.y = Arow[kb + 1];
    const int k0 = kb + 2 * hh;
    const int k1 = k0 + 1;
    const float* r0 = ((k0 < CH) ? (x + (long)k0 * HW) : (xt + (long)(k0 - CH) * HW)) + n0 + ml;
    const float* r1 = ((k1 < CH) ? (x + (long)k1 * HW) : (xt + (long)(k1 - CH) * HW)) + n0 + ml;
    v2f b0 = { r0[0],  r1[0]  };
    v2f b1 = { r0[16], r1[16] };
    v2f b2 = { r0[32], r1[32] };
    v2f b3 = { r0[48], r1[48] };
    acc0 = wmma4(a, b0, acc0);
    acc1 = wmma4(a, b1, acc1);
    acc2 = wmma4(a, b2, acc2);
    acc3 = wmma4(a, b3, acc3);
  }
#pragma unroll
  for (int r = 0; r < 8; ++r) {
    const int o = m0 + r + 8 * hh;
    const float sc  = gamma[o] * rsqrtf(rvar[o] + 1e-5f);
    const float bb  = beta[o];
    const float mu  = rmean[o];
    const float bvv = bv[o];
    float* yp = y + (long)o * HW + n0 + ml;
    yp[0]  = fmaxf((acc0[r] + bvv - mu) * sc + bb, 0.f);
    yp[16] = fmaxf((acc1[r] + bvv - mu) * sc + bb, 0.f);
    yp[32] = fmaxf((acc2[r] + bvv - mu) * sc + bb, 0.f);
    yp[48] = fmaxf((acc3[r] + bvv - mu) * sc + bb, 0.f);
  }
}

extern "C" void kernel_launch(void* const* d_in, const int* in_sizes, int n_in,
                              void* d_out, int out_size, void* d_ws, size_t ws_size,
                              hipStream_t stream) {
  (void)in_sizes; (void)n_in; (void)out_size; (void)ws_size;
  const float* x     = (const float*)d_in[0];
  const float* Wq    = (const float*)d_in[1];
  const float* Wk    = (const float*)d_in[2];
  const float* Wv    = (const float*)d_in[3];
  const float* bv    = (const float*)d_in[4];
  const float* gamma = (const float*)d_in[5];
  const float* beta  = (const float*)d_in[6];
  const float* rmean = (const float*)d_in[7];
  const float* rvar  = (const float*)d_in[8];
  // d_in[9] = k (== 7, compile-time assumed)
  float* out = (float*)d_out;

  float* ws = (float*)d_ws;
  const long qSz = (long)CQ * HW;   // per-image 64-d map
  const long cSz = (long)CH * HW;   // per-image 256-d feature map
  float* Q   = ws;                  // 2 * qSz
  float* Ks  = Q  + 2 * qSz;        // 2 * qSz
  float* Km  = Ks + 2 * qSz;        // 2 * qSz
  float* XT  = Km + 2 * qSz;        // 4 * cSz  (xt1 imgs 0,1 | xt2 imgs 2,3)
  float* xt1 = XT;
  float* xt2 = XT + 2 * cSz;

  const float* x12 = x;             // images 0,1 (original xt1)
  const float* x34 = x + 2 * cSz;   // images 2,3 (original xt2)

  const dim3 blk(256, 1, 1);
  const dim3 gQ(25, 1, 2);          // ceil(4*49 waves / 8 waves-per-block)
  const dim3 gA(HW / 8, 1, 2);      // wave per pixel
  const dim3 gF(98, 1, 4);          // 16*49 waves / 8

  // Phase 1: xt1_new = align(ta = x[2:4], tb = x[0:2])
  gemm_wmma<<<gQ, blk, 0, stream>>>(Wq, x34, Q,  CQ, CH, cSz, qSz);
  gemm_wmma<<<gQ, blk, 0, stream>>>(Wk, x34, Ks, CQ, CH, cSz, qSz);
  gemm_wmma<<<gQ, blk, 0, stream>>>(Wk, x12, Km, CQ, CH, cSz, qSz);
  attn_kernel<<<gA, blk, 0, stream>>>(Q, Ks, Km, x12, xt1, qSz, cSz);

  // Phase 2: xt2_new = align(ta = xt1_new, tb = x[2:4])
  gemm_wmma<<<gQ, blk, 0, stream>>>(Wq, xt1, Q,  CQ, CH, cSz, qSz);
  gemm_wmma<<<gQ, blk, 0, stream>>>(Wk, xt1, Ks, CQ, CH, cSz, qSz);
  gemm_wmma<<<gQ, blk, 0, stream>>>(Wk, x34, Km, CQ, CH, cSz, qSz);
  attn_kernel<<<gA, blk, 0, stream>>>(Q, Ks, Km, x34, xt2, qSz, cSz);

  // Final: Wv * [x;xt] + bv -> BN -> ReLU
  final_wmma<<<gF, blk, 0, stream>>>(Wv, x, XT, bv, gamma, beta, rmean, rvar, out);
}